// ConvSelfLayer_new_29240137351515
// MI455X (gfx1250) — compile-verified
//
#include <hip/hip_runtime.h>
#include <hip/hip_bf16.h>

// CDNA5 / gfx1250: per-row full self-convolution via fp32 WMMA (16x16x4).
// out[b, o] = sum_j x[b,j] * x[b, o-j],  o in [0, 2N-2],  N=512.
//
// Toeplitz-tiled GEMM: for output super-tile o = tileBase + 16c + t,
//   D[t,c] = sum_u A[t,u]*B[u,c],  A[t,u]=x[tileBase+t-u], B[u,c]=x[16c+u]
// Chained V_WMMA_F32_16X16X4_F32, fp32-exact math.
//
// v4: B-fragment loads are explicit 8B-aligned v2f loads (boff is always
//     even) -> ds_load_b64/2addr_b64 forms; A stays as scalar pairs (parity
//     of its address flips with t). Group-of-8 body, no remainder loop,
//     scalar trip count (EXEC all-ones at every WMMA).

typedef __attribute__((ext_vector_type(2))) float v2f;
typedef __attribute__((ext_vector_type(8))) float v8f;

#define NLEN   512
#define OLEN   1023          // 2N-1
#define PAD    256
#define XPAD   1024          // padded row length; all indices stay in [0,1024)

__global__ __launch_bounds__(256)
void conv_self_wmma_f32_kernel(const float* __restrict__ x,
                               float* __restrict__ out,
                               int batch) {
    // LDS per row: fwd padded copy [0,1024) + reversed copy [1024,2048).
    // 2 rows (16 KB) + 8 output tiles (8 KB) = 24 KB.
    __shared__ float lds[2 * 2 * XPAD + 8 * 256];

    const int tid  = threadIdx.x;
    const int lane = tid & 31;
    const int wave = tid >> 5;              // 0..7
    const int lrow = wave >> 2;             // 0..1 : row within block
    const int tileBase = (wave & 3) << 8;   // 0,256,512,768 : output tile base

    const int row0 = blockIdx.x * 2;

    // ---- stage 2 rows into LDS, zero-padded, fwd + reversed copies
    for (int idx = tid; idx < 2 * XPAD; idx += 256) {
        const int r = idx >> 10;            // 0..1
        const int i = idx & (XPAD - 1);     // 0..1023
        const int g = i - PAD;              // source index into the row
        float v = 0.0f;
        if (g >= 0 && g < NLEN) {
            v = x[(size_t)(row0 + r) * NLEN + g];
        }
        const int base = r * 2 * XPAD;
        lds[base + i] = v;                          // xpad[i]
        lds[base + XPAD + (XPAD - 1 - i)] = v;      // xrev[j] = xpad[1023-j]
    }
    __syncthreads();

    const float* __restrict__ xf = &lds[lrow * 2 * XPAD];         // fwd
    const float* __restrict__ xv = &lds[lrow * 2 * XPAD + XPAD];  // rev
    float* __restrict__ tile = &lds[2 * 2 * XPAD + wave * 256];

    const int t  = lane & 15;   // M index (and B column index)
    const int kh = lane >> 4;   // lane half: K slot = 2*kh + vgpr

    // u-window where both operands can be nonzero
    int u_lo = tileBase - (NLEN - 1);
    if (u_lo < -(15 * 16)) u_lo = -(15 * 16);   // max(-240, tileBase-511)
    int u_hi = tileBase + 15;
    if (u_hi > NLEN - 1) u_hi = NLEN - 1;       // min(511, tileBase+15)
    const int u_start = u_lo & ~3;              // align to K=4 chunk

    // Groups of 8 K=4 chunks (32 u-values): 8/16/16/8 per tile -> exact.
    // Wave-uniform; force into SGPR so the loop is a scalar branch and EXEC
    // stays all-ones (WMMA requirement). No remainder loop exists.
    const int nGroups =
        __builtin_amdgcn_readfirstlane((u_hi + 1 - u_start) >> 5);

    // A[t,u] = xpad[tileBase + t - u + PAD] = xrev[1023 - (tileBase+t-u+PAD)]
    //   slot k = 2kh+r  ->  a[r] = ap[4*chunk + r], ascending (parity varies
    //   with t -> scalar pair loads, merged into ds_load_2addr_b32)
    const float* __restrict__ ap =
        xv + ((XPAD - 1 - PAD) - tileBase - t + 2 * kh + u_start);
    // B[u,c] = xpad[16c + u + PAD], c = t -> b[r] = bp[4*chunk + r]
    // boff = 16t + 256 + 2kh + u_start: every term even -> 8B aligned.
    const float* __restrict__ bp =
        xf + (16 * t + PAD + 2 * kh + u_start);

    v8f acc = {};
    for (int g = 0; g < nGroups; ++g) {
#pragma unroll
        for (int j = 0; j < 8; ++j) {
            v2f a;
            a.x = ap[4 * j];
            a.y = ap[4 * j + 1];
            const v2f b = *(const v2f*)(bp + 4 * j);   // aligned b64 load
            acc = __builtin_amdgcn_wmma_f32_16x16x4_f32(
                /*neg_a=*/false, a, /*neg_b=*/false, b,
                /*c_mod=*/(short)0, acc, /*reuse_a=*/false, /*reuse_b=*/false);
        }
        ap += 32;
        bp += 32;
    }

    // ---- stage D through LDS for coalesced stores.
    // D layout: VGPR r, lanes 0-15 -> M=r, N=lane; lanes 16-31 -> M=r+8
    // output offset within super-tile: j = 16*N + M  (o = tileBase + j)
#pragma unroll
    for (int r = 0; r < 8; ++r) {
        const int M = r + 8 * kh;
        tile[16 * t + M] = acc[r];
    }
    __syncthreads();

    // ---- coalesced writeback; clip o < 1023
    const size_t obase = (size_t)(row0 + lrow) * OLEN + tileBase;
#pragma unroll
    for (int j = lane; j < 256; j += 32) {
        if (tileBase + j < OLEN) {
            out[obase + j] = tile[j];
        }
    }
}

extern "C" void kernel_launch(void* const* d_in, const int* in_sizes, int n_in,
                              void* d_out, int out_size, void* d_ws, size_t ws_size,
                              hipStream_t stream) {
    const float* x = (const float*)d_in[0];
    float* out = (float*)d_out;
    const int batch = in_sizes[0] / NLEN;     // 16384
    const int blocks = batch / 2;             // 2 rows per 256-thread block
    conv_self_wmma_f32_kernel<<<blocks, 256, 0, stream>>>(x, out, batch);
}